// ColorGNNEmbedding_72748156060190
// MI455X (gfx1250) — compile-verified
//
#include <hip/hip_runtime.h>

// Problem constants (from reference)
#define NN   20000          // nodes
#define EE   320000         // edges
#define XW   1005           // x row width: 1 + 1000 + 1 + 3
#define FS   1000           // resnet features
#define HF   1755           // embedded width: 250 + 1000 + 250 + 255
#define HFP  1760           // HF padded to multiple of 32 (16B-aligned rows)
#define F1   512
#define F2   256
#define F3   64
#define MTILE 128
#define MP   20096          // NN padded to multiple of MTILE (157 * 128)

typedef __attribute__((ext_vector_type(2))) float v2f;
typedef __attribute__((ext_vector_type(8))) float v8f;

// ---------------------------------------------------------------- degree
__global__ void k_deg_init(float* deg) {
    int i = blockIdx.x * blockDim.x + threadIdx.x;
    if (i < NN) deg[i] = 1.0f;                 // segment_sum(ew) + 1.0
}

__global__ void k_deg_scatter(const int* __restrict__ dst, const float* __restrict__ ew,
                              float* deg) {
    int e = blockIdx.x * blockDim.x + threadIdx.x;
    if (e < EE) atomicAdd(&deg[dst[e]], ew[e]);
}

__global__ void k_dinv(float* deg_dinv) {
    int i = blockIdx.x * blockDim.x + threadIdx.x;
    if (i < NN) {
        float d = deg_dinv[i];
        deg_dinv[i] = d > 0.0f ? rsqrtf(d) : 0.0f;
    }
}

// ---------------------------------------------------------------- embedding (row stride HFP, pad = 0)
__global__ void k_embed(const float* __restrict__ x,
                        const float* __restrict__ layer_table,
                        const float* __restrict__ rel_table,
                        const float* __restrict__ color_table,
                        float* __restrict__ H) {
    int n = blockIdx.x;
    float* hr = H + (size_t)n * HFP;
    if (n >= NN) {                              // M-pad rows: zero
        for (int f = threadIdx.x; f < HFP; f += blockDim.x) hr[f] = 0.0f;
        return;
    }
    const float* xr = x + (size_t)n * XW;
    int li = (int)(xr[0] - 1.0f);                       // 0..2
    int ri = (int)lrintf(fabsf(xr[FS + 1]) * 10.0f);    // 0..10, RNE like jnp.round
    int c0 = (int)xr[XW - 3], c1 = (int)xr[XW - 2], c2 = (int)xr[XW - 1];
    for (int f = threadIdx.x; f < HFP; f += blockDim.x) {
        float v = 0.0f;
        if (f < 250)        v = layer_table[li * 250 + f];
        else if (f < 1250)  v = xr[1 + (f - 250)];
        else if (f < 1500)  v = rel_table[ri * 250 + (f - 1250)];
        else if (f < HF) {
            int q = f - 1500, c = q / 85, j = q - c * 85;
            int ci = (c == 0) ? c0 : ((c == 1) ? c1 : c2);
            v = color_table[ci * 85 + j];
        }
        hr[f] = v;
    }
}

// ---------------------------------------------------------------- fp32 WMMA GEMM
// C[MP x Ncol] = act(A)[MP x Kpad] @ W[Kreal x Ncol]
// block = 256 threads (8 waves); block tile 128x64; wave tile 32x32 (4 accumulators);
// K staged through LDS 32 at a time with float4 (b128) global->LDS staging.
// No M/K guards on A: caller pads A to MP rows and Kpad cols (Kpad % 32 == 0).
template<bool LRELU>
__global__ void __launch_bounds__(256)
k_gemm_wmma(const float* __restrict__ A, int lda, int Kpad, int Kreal,
            const float* __restrict__ W, float* __restrict__ C, int Ncol) {
    __shared__ float As[MTILE][36];   // 128x32 tile, stride 36 (16B aligned, conflict-free)
    __shared__ float Bs[32][72];      // 32x64 tile, stride 72

    const int t     = threadIdx.x;
    const int lane  = t & 31;
    const int w     = t >> 5;             // 0..7
    const int wm    = (w >> 1) << 5;      // wave tile row: 0,32,64,96
    const int wn    = (w & 1) << 5;       // wave tile col: 0,32
    const int l16   = lane & 15;
    const int khalf = (lane >> 4) << 1;   // 0 | 2

    const int blockM = blockIdx.x * MTILE;
    const int blockN = blockIdx.y * 64;

    v8f acc00 = {}, acc01 = {}, acc10 = {}, acc11 = {};

    for (int k0 = 0; k0 < Kpad; k0 += 32) {
        // ---- stage A tile 128x32 (1024 float4, 4 per thread), lrelu fused on read
        #pragma unroll
        for (int j = 0; j < 4; ++j) {
            int i  = t + j * 256;         // 0..1023
            int r  = i >> 3;              // 0..127
            int c4 = (i & 7) << 2;        // 0,4,..,28
            float4 v = *(const float4*)(A + (size_t)(blockM + r) * lda + k0 + c4);
            if (LRELU) {
                v.x = v.x > 0.0f ? v.x : 0.01f * v.x;
                v.y = v.y > 0.0f ? v.y : 0.01f * v.y;
                v.z = v.z > 0.0f ? v.z : 0.01f * v.z;
                v.w = v.w > 0.0f ? v.w : 0.01f * v.w;
            }
            *(float4*)(&As[r][c4]) = v;
        }
        // ---- stage B tile 32x64 (512 float4, 2 per thread); K-guard only vs Kreal
        #pragma unroll
        for (int j = 0; j < 2; ++j) {
            int i  = t + j * 256;         // 0..511
            int r  = i >> 4;              // 0..31
            int c4 = (i & 15) << 2;       // 0,4,..,60
            int bk = k0 + r;
            float4 v = make_float4(0.0f, 0.0f, 0.0f, 0.0f);
            if (bk < Kreal) v = *(const float4*)(W + (size_t)bk * Ncol + blockN + c4);
            *(float4*)(&Bs[r][c4]) = v;
        }
        __syncthreads();

        // ---- 8 WMMA k-steps of K=4; 4 independent accumulator chains
        #pragma unroll
        for (int kk = 0; kk < 8; ++kk) {
            int kb = kk * 4 + khalf;
            v2f a0, a1, b0, b1;
            a0.x = As[wm + l16][kb];        a0.y = As[wm + l16][kb + 1];
            a1.x = As[wm + 16 + l16][kb];   a1.y = As[wm + 16 + l16][kb + 1];
            b0.x = Bs[kb][wn + l16];        b0.y = Bs[kb + 1][wn + l16];
            b1.x = Bs[kb][wn + 16 + l16];   b1.y = Bs[kb + 1][wn + 16 + l16];
            acc00 = __builtin_amdgcn_wmma_f32_16x16x4_f32(false, a0, false, b0, (short)0, acc00, false, false);
            acc01 = __builtin_amdgcn_wmma_f32_16x16x4_f32(false, a0, false, b1, (short)0, acc01, false, false);
            acc10 = __builtin_amdgcn_wmma_f32_16x16x4_f32(false, a1, false, b0, (short)0, acc10, false, false);
            acc11 = __builtin_amdgcn_wmma_f32_16x16x4_f32(false, a1, false, b1, (short)0, acc11, false, false);
        }
        __syncthreads();
    }

    // ---- store: D layout VGPR r -> M = r (lanes 0-15) / r+8 (lanes 16-31); N = lane&15
    const int rofs = (lane >> 4) << 3;
    #pragma unroll
    for (int r = 0; r < 8; ++r) {
        size_t row0 = (size_t)(blockM + wm + r + rofs);
        size_t row1 = row0 + 16;
        C[row0 * Ncol + blockN + wn + l16]      = acc00[r];
        C[row0 * Ncol + blockN + wn + 16 + l16] = acc01[r];
        C[row1 * Ncol + blockN + wn + l16]      = acc10[r];
        C[row1 * Ncol + blockN + wn + 16 + l16] = acc11[r];
    }
}

// ---------------------------------------------------------------- aggregation
__global__ void k_agg_init(const float* __restrict__ hp, const float* __restrict__ dinv,
                           const float* __restrict__ b, float* __restrict__ agg,
                           int total, int F) {
    int i = blockIdx.x * blockDim.x + threadIdx.x;
    if (i < total) {
        int n = i / F, f = i - n * F;
        float di = dinv[n];
        agg[i] = hp[i] * di * di + b[f];
    }
}

// Each thread handles 4 features of one edge; 2^qshift threads per edge.
__global__ void k_agg_scatter(const float* __restrict__ hp, const int* __restrict__ src,
                              const int* __restrict__ dst, const float* __restrict__ ew,
                              const float* __restrict__ dinv, float* __restrict__ agg,
                              int F, int qshift) {
    int t  = threadIdx.x;
    int le = t >> qshift;                         // local edge within block
    int q  = (t & ((1 << qshift) - 1)) << 2;      // feature offset (x4)
    int e  = blockIdx.x * (blockDim.x >> qshift) + le;
    if (e >= EE) return;
    int s = src[e], d = dst[e];
    float norm = dinv[s] * ew[e] * dinv[d];
    float4 h = *(const float4*)(hp + (size_t)s * F + q);
    float* ad = agg + (size_t)d * F + q;
    atomicAdd(ad + 0, h.x * norm);
    atomicAdd(ad + 1, h.y * norm);
    atomicAdd(ad + 2, h.z * norm);
    atomicAdd(ad + 3, h.w * norm);
}

// ---------------------------------------------------------------- final projection
__global__ void k_final(const float* __restrict__ agg, const float* __restrict__ Wp,
                        const float* __restrict__ bp, float* __restrict__ out, int M) {
    int n = blockIdx.x * blockDim.x + threadIdx.x;
    if (n >= M) return;
    const float* a = agg + (size_t)n * F3;
    float o0 = bp[0], o1 = bp[1], o2 = bp[2];
    #pragma unroll 8
    for (int k = 0; k < F3; ++k) {
        float v = a[k];
        v = v > 0.0f ? v : 0.01f * v;   // lrelu(h3)
        o0 += v * Wp[k * 3 + 0];
        o1 += v * Wp[k * 3 + 1];
        o2 += v * Wp[k * 3 + 2];
    }
    out[n * 3 + 0] = o0;
    out[n * 3 + 1] = o1;
    out[n * 3 + 2] = o2;
}

// ---------------------------------------------------------------- launch
extern "C" void kernel_launch(void* const* d_in, const int* in_sizes, int n_in,
                              void* d_out, int out_size, void* d_ws, size_t ws_size,
                              hipStream_t stream) {
    const float* x           = (const float*)d_in[0];
    const int*   edge_index  = (const int*)  d_in[1];   // (2,E): src then dst
    const float* edge_attr   = (const float*)d_in[2];
    const float* layer_table = (const float*)d_in[3];
    const float* rel_table   = (const float*)d_in[4];
    const float* color_table = (const float*)d_in[5];
    const float* W1 = (const float*)d_in[6];
    const float* b1 = (const float*)d_in[7];
    const float* W2 = (const float*)d_in[8];
    const float* b2 = (const float*)d_in[9];
    const float* W3 = (const float*)d_in[10];
    const float* b3 = (const float*)d_in[11];
    const float* Wp = (const float*)d_in[12];
    const float* bp = (const float*)d_in[13];
    float* out = (float*)d_out;

    const int* src = edge_index;
    const int* dst = edge_index + EE;

    // workspace carve-out (256B aligned); M-padded to MP rows everywhere
    size_t off = 0;
    auto carve = [&](size_t bytes) {
        size_t p = off;
        off = (off + bytes + 255) & ~(size_t)255;
        return p;
    };
    float* H    = (float*)((char*)d_ws + carve((size_t)MP * HFP * 4)); // 141.5 MB
    float* bufA = (float*)((char*)d_ws + carve((size_t)MP * F1 * 4)); //  41.2 MB (h')
    float* bufB = (float*)((char*)d_ws + carve((size_t)MP * F1 * 4)); //  41.2 MB (agg)
    float* dinv = (float*)((char*)d_ws + carve((size_t)NN * 4));
    (void)ws_size; (void)n_in; (void)in_sizes; (void)out_size;

    // degree -> dinv (in place)
    k_deg_init   <<<(NN + 255) / 256, 256, 0, stream>>>(dinv);
    k_deg_scatter<<<(EE + 255) / 256, 256, 0, stream>>>(dst, edge_attr, dinv);
    k_dinv       <<<(NN + 255) / 256, 256, 0, stream>>>(dinv);

    // embedding (zero-fills K-pad cols and M-pad rows)
    k_embed<<<MP, 256, 0, stream>>>(x, layer_table, rel_table, color_table, H);

    dim3 blk(256);
    const int mg = MP / MTILE;            // 157

    // layer 1: h' = H @ W1 ; agg1 = scatter + self + b1  (lrelu deferred to next A-read)
    k_gemm_wmma<false><<<dim3(mg, F1 / 64), blk, 0, stream>>>(H, HFP, HFP, HF, W1, bufA, F1);
    k_agg_init  <<<((NN * F1) + 255) / 256, 256, 0, stream>>>(bufA, dinv, b1, bufB, NN * F1, F1);
    k_agg_scatter<<<(EE + 1) / 2, 256, 0, stream>>>(bufA, src, dst, edge_attr, dinv, bufB, F1, 7);

    // layer 2: h' = lrelu(agg1) @ W2
    k_gemm_wmma<true><<<dim3(mg, F2 / 64), blk, 0, stream>>>(bufB, F1, F1, F1, W2, bufA, F2);
    k_agg_init  <<<((NN * F2) + 255) / 256, 256, 0, stream>>>(bufA, dinv, b2, bufB, NN * F2, F2);
    k_agg_scatter<<<(EE + 3) / 4, 256, 0, stream>>>(bufA, src, dst, edge_attr, dinv, bufB, F2, 6);

    // layer 3: h' = lrelu(agg2) @ W3
    k_gemm_wmma<true><<<dim3(mg, F3 / 64), blk, 0, stream>>>(bufB, F2, F2, F2, W3, bufA, F3);
    k_agg_init  <<<((NN * F3) + 255) / 256, 256, 0, stream>>>(bufA, dinv, b3, bufB, NN * F3, F3);
    k_agg_scatter<<<(EE + 15) / 16, 256, 0, stream>>>(bufA, src, dst, edge_attr, dinv, bufB, F3, 4);

    // out = lrelu(agg3) @ Wp + bp
    k_final<<<(NN + 255) / 256, 256, 0, stream>>>(bufB, Wp, bp, out, NN);
}